// TopkPointExtractor_3152505995766
// MI455X (gfx1250) — compile-verified
//
#include <hip/hip_runtime.h>
#include <hip/hip_bf16.h>
#include <math.h>
#include <stdint.h>

// ---------------------------------------------------------------------------
// Problem constants (from the reference)
// ---------------------------------------------------------------------------
#define BATCH   16
#define C_IN    512
#define HDIM    64
#define WDIM    64
#define NPIX    (HDIM * WDIM)      // 4096
#define ENC     64
#define P_TOP   256
#define OUTF    (3 + ENC)          // 67

typedef __attribute__((ext_vector_type(2))) float v2f;
typedef __attribute__((ext_vector_type(8))) float v8f;

// ---------------------------------------------------------------------------
// CDNA5 async global->LDS copy (GLOBAL_LOAD_ASYNC_TO_LDS_B128, ASYNCcnt).
// The ISA (§10.2) maps generic LDS-aperture addresses to the wave's LDS by
// truncating to addr[31:0], so the low 32 bits of a generic pointer to a
// __shared__ object are the LDS byte address the instruction expects in vdst.
// ---------------------------------------------------------------------------
__device__ __forceinline__ void async_b128_to_lds(const float* gsrc, float* ldst)
{
    uint32_t lds_addr = (uint32_t)(uintptr_t)ldst;
    uint64_t gaddr    = (uint64_t)(uintptr_t)gsrc;
    asm volatile("global_load_async_to_lds_b128 %0, %1, off"
                 :: "v"(lds_addr), "v"(gaddr) : "memory");
}

__device__ __forceinline__ void wait_async0()
{
#if __has_builtin(__builtin_amdgcn_s_wait_asynccnt)
    __builtin_amdgcn_s_wait_asynccnt(0);
#else
    asm volatile("s_wait_asynccnt 0" ::: "memory");
#endif
}

// ---------------------------------------------------------------------------
// Kernel 1: batched GEMM (1x1 conv) via V_WMMA_F32_16X16X4_F32 + bias,
// fused per-pixel sum-of-squares reduction.
//   grid  = (NPIX/64, BATCH), block = 128 threads (4 wave32s)
//   wave m = M-tile m*16..m*16+15 (ENC rows), block covers 64 N columns.
// Data movement: img K-chunks (32 x 64 f32 = 8KB) double-buffered in LDS via
// async global->LDS DMA (no VGPR round trip), shared by all 4 waves; w
// fragments read direct (L2-resident 128KB); prefetch one chunk further out.
// ---------------------------------------------------------------------------
__global__ __launch_bounds__(128)
void topk_gemm_kernel(const float* __restrict__ img,   // [B, C_IN, NPIX]
                      const float* __restrict__ wmat,  // [ENC, C_IN]
                      const float* __restrict__ bias,  // [ENC]
                      float* __restrict__ xfull,       // [B, ENC, NPIX]
                      float* __restrict__ xout)        // [B, NPIX]
{
    const int KCHUNK = 32;
    __shared__ __align__(16) float ldsB[2][KCHUNK * 64];  // [buf][k][col], 16 KB
    __shared__ float ldsX[64];                             // per-column sum(x^2)

    const int tid   = threadIdx.x;
    const int lane  = tid & 31;
    const int wave  = tid >> 5;           // 0..3  -> M tile
    const int b     = blockIdx.y;
    const int nBase = blockIdx.x * 64;    // N tile base (64 columns)

    const int mlane = lane & 15;          // M index (A) / N index (B,C,D)
    const int kb    = (lane >> 4) * 2;    // K sub-offset within step of 4
    const int mBase = wave * 16;

    if (tid < 64) ldsX[tid] = 0.0f;

    v8f acc[4] = {};                      // 4 N-subtiles of 16x16 f32

    const float* imgB = img + ((size_t)b * C_IN) * NPIX + nBase;

    // Per-thread slice of a chunk: 4 x b128 (2048 floats / 128 threads)
    const int s_f4  = tid;                            // base float4 id
    // issue async DMA for chunk starting at k0 into buffer `buf`
    auto issue_chunk = [&](int k0, int buf) {
        #pragma unroll
        for (int i = 0; i < 4; ++i) {
            int f4  = s_f4 + i * 128;          // 0..511
            int kk  = f4 >> 4;                 // 0..31
            int col = (f4 & 15) << 2;          // 0..60 step 4
            async_b128_to_lds(imgB + ((size_t)(k0 + kk)) * NPIX + col,
                              &ldsB[buf][kk * 64 + col]);
        }
        // warm L2 one chunk further out (global_prefetch_b8)
        if (k0 + KCHUNK < C_IN)
            __builtin_prefetch(imgB + ((size_t)(k0 + KCHUNK + (tid & 31))) * NPIX, 0, 1);
    };

    issue_chunk(0, 0);

    for (int k0 = 0; k0 < C_IN; k0 += KCHUNK) {
        const int buf = (k0 / KCHUNK) & 1;
        wait_async0();          // this wave's DMA into `buf` done
        __syncthreads();        // all waves' DMA visible; prior readers of
                                // buf^1 are past their loads
        if (k0 + KCHUNK < C_IN)
            issue_chunk(k0 + KCHUNK, buf ^ 1);

        const float* lb = &ldsB[buf][0];
        #pragma unroll
        for (int kk = 0; kk < KCHUNK; kk += 4) {
            // A fragment: 16x4 f32.  lanes 0-15: K=+0,+1 ; lanes 16-31: K=+2,+3
            const float2 aw = *(const float2*)(
                wmat + (size_t)(mBase + mlane) * C_IN + (k0 + kk + kb));
            v2f a; a.x = aw.x; a.y = aw.y;
            #pragma unroll
            for (int nt = 0; nt < 4; ++nt) {
                const int col = nt * 16 + mlane;
                v2f bf;
                bf.x = lb[(kk + kb)     * 64 + col];
                bf.y = lb[(kk + kb + 1) * 64 + col];
                acc[nt] = __builtin_amdgcn_wmma_f32_16x16x4_f32(
                    /*neg_a=*/false, a, /*neg_b=*/false, bf,
                    /*c_mod=*/(short)0, acc[nt],
                    /*reuse_a=*/false, /*reuse_b=*/false);
            }
        }
    }
    __syncthreads();

    // Epilogue: bias add, store x_full, fused sum-of-squares per column.
    // C/D layout: lanes 0-15 -> M = r (VGPR r), N = lane;
    //             lanes 16-31 -> M = 8 + r,     N = lane - 16.
    const int msel = (lane >> 4) * 8;
    float bv[8];
    #pragma unroll
    for (int r = 0; r < 8; ++r) bv[r] = bias[mBase + msel + r];

    #pragma unroll
    for (int nt = 0; nt < 4; ++nt) {
        const int col = nBase + nt * 16 + mlane;
        float ss = 0.0f;
        #pragma unroll
        for (int r = 0; r < 8; ++r) {
            const int row = mBase + msel + r;
            const float v = acc[nt][r] + bv[r];
            xfull[((size_t)b * ENC + row) * NPIX + col] = v;
            ss += v * v;
        }
        // combine the two half-lanes holding the same column (M halves)
        ss += __shfl_xor(ss, 16);
        if (lane < 16) atomicAdd(&ldsX[nt * 16 + mlane], ss);  // ds_add_f32
    }
    __syncthreads();
    if (tid < 64)
        xout[(size_t)b * NPIX + nBase + tid] = ldsX[tid];
}

// ---------------------------------------------------------------------------
// Kernel 2: exact top-256 per batch (sorted descending, lowest-index ties),
// iterative argmax over an LDS-resident copy.  grid = BATCH, block = 256.
// ---------------------------------------------------------------------------
__global__ __launch_bounds__(256)
void topk_select_kernel(const float* __restrict__ x, int* __restrict__ inds)
{
    __shared__ float vals[NPIX];     // 16 KB
    __shared__ float rv[256];
    __shared__ int   ri[256];

    const int b = blockIdx.x, tid = threadIdx.x;
    for (int i = tid; i < NPIX; i += 256)
        vals[i] = x[(size_t)b * NPIX + i];
    __syncthreads();

    for (int it = 0; it < P_TOP; ++it) {
        float best = -INFINITY; int bi = NPIX;
        for (int i = tid; i < NPIX; i += 256) {
            const float v = vals[i];
            if (v > best) { best = v; bi = i; }
        }
        rv[tid] = best; ri[tid] = bi;
        __syncthreads();
        for (int s = 128; s > 0; s >>= 1) {
            if (tid < s) {
                const float ov = rv[tid + s]; const int oi = ri[tid + s];
                if (ov > rv[tid] || (ov == rv[tid] && oi < ri[tid])) {
                    rv[tid] = ov; ri[tid] = oi;
                }
            }
            __syncthreads();
        }
        if (tid == 0) {
            inds[b * P_TOP + it] = ri[0];
            vals[ri[0]] = -INFINITY;
        }
        __syncthreads();
    }
}

// ---------------------------------------------------------------------------
// Kernel 3: feature gather + 5x5 soft-coordinate refiner at selected points.
//   grid = BATCH*P_TOP, block = 64 (one lane per ENC channel).
// kern0 = ordKer*sw, kern1 = absKer*sw, kern2 = sw with sw = 5-(|ri|+|rj|);
// XLA conv is cross-correlation with zero padding 2.
// ---------------------------------------------------------------------------
__global__ __launch_bounds__(64)
void refine_gather_kernel(const float* __restrict__ xfull,
                          const float* __restrict__ x,
                          const int*   __restrict__ inds,
                          float*       __restrict__ out)   // [B, P, 67]
{
    const int bp  = blockIdx.x;
    const int b   = bp >> 8;
    const int t   = threadIdx.x;
    const int ind = inds[bp];

    float* orow = out + (size_t)bp * OUTF;
    orow[3 + t] = xfull[((size_t)b * ENC + t) * NPIX + ind];

    if (t == 0) {
        const int ordc = ind >> 6;       // row (H)
        const int absc = ind & 63;       // col (W)
        float s_ord = 0.0f, s_abs = 0.0f, s_w = 0.0f;
        #pragma unroll
        for (int i = 0; i < 5; ++i) {
            #pragma unroll
            for (int j = 0; j < 5; ++j) {
                const int yy = ordc + i - 2, xx = absc + j - 2;
                if (yy >= 0 && yy < HDIM && xx >= 0 && xx < WDIM) {
                    const float v  = x[(size_t)b * NPIX + (yy << 6) + xx];
                    const float ri = (float)(i - 2), rj = (float)(j - 2);
                    const float sw = 5.0f - fabsf(ri) - fabsf(rj);
                    s_ord += v * ri * sw;
                    s_abs += v * rj * sw;
                    s_w   += v * sw;
                }
            }
        }
        const float inv = 1.0f / (s_w + 0.001f);
        orow[0] = (float)absc + s_abs * inv;   // absf
        orow[1] = (float)ordc + s_ord * inv;   // ordf
        orow[2] = 0.0f;
    }
}

// ---------------------------------------------------------------------------
// Launch
// ---------------------------------------------------------------------------
extern "C" void kernel_launch(void* const* d_in, const int* in_sizes, int n_in,
                              void* d_out, int out_size, void* d_ws, size_t ws_size,
                              hipStream_t stream)
{
    const float* img  = (const float*)d_in[0];   // [16,512,64,64]
    const float* wmat = (const float*)d_in[1];   // [64,512]
    const float* bias = (const float*)d_in[2];   // [64]
    float* out = (float*)d_out;                  // [16,256,67]

    // Workspace layout
    float* xfull = (float*)d_ws;                               // 16*64*4096 f32
    float* x     = xfull + (size_t)BATCH * ENC * NPIX;         // 16*4096 f32
    int*   inds  = (int*)(x + (size_t)BATCH * NPIX);           // 16*256 i32

    dim3 g1(NPIX / 64, BATCH);
    topk_gemm_kernel<<<g1, 128, 0, stream>>>(img, wmat, bias, xfull, x);

    topk_select_kernel<<<BATCH, 256, 0, stream>>>(x, inds);

    refine_gather_kernel<<<BATCH * P_TOP, 64, 0, stream>>>(xfull, x, inds, out);
}